// LmHeadWithSampleHead_39333310497459
// MI455X (gfx1250) — compile-verified
//
#include <hip/hip_runtime.h>
#include <hip/hip_bf16.h>
#include <math.h>

typedef __attribute__((ext_vector_type(2))) float v2f;
typedef __attribute__((ext_vector_type(4))) float v4f;
typedef __attribute__((ext_vector_type(8))) float v8f;

#define BATCH 64
#define DIM   4096
#define VOCAB 32000
#define TOPK  50
#define MINKEEP 5
#define BK    32            // K-chunk staged in LDS
#define LDS_STRIDE (BK + 4) // pad: 36 floats -> 144B row stride, conflict-free

// ---------------- LayerNorm: x = (h-mu)*rsqrt(var+eps)*w + b ----------------
__global__ __launch_bounds__(256)
void ln_kernel(const float* __restrict__ hs, const float* __restrict__ w,
               const float* __restrict__ b, float* __restrict__ out) {
    __shared__ float red[256];
    const int row = blockIdx.x, tid = threadIdx.x;
    const float* hr = hs + row * DIM;
    float s = 0.f, s2 = 0.f;
    for (int i = tid; i < DIM; i += 256) { float v = hr[i]; s += v; s2 += v * v; }
    red[tid] = s; __syncthreads();
    for (int st = 128; st > 0; st >>= 1) { if (tid < st) red[tid] += red[tid + st]; __syncthreads(); }
    const float mean = red[0] * (1.0f / DIM);
    __syncthreads();
    red[tid] = s2; __syncthreads();
    for (int st = 128; st > 0; st >>= 1) { if (tid < st) red[tid] += red[tid + st]; __syncthreads(); }
    const float var = red[0] * (1.0f / DIM) - mean * mean;
    const float rstd = rsqrtf(var + 1e-5f);
    float* orow = out + row * DIM;
    for (int i = tid; i < DIM; i += 256)
        orow[i] = (hr[i] - mean) * rstd * w[i] + b[i];
}

// ------------- GEMM: logits[64,V] = x[64,D] * lm_w[V,D]^T via WMMA f32 -------------
// grid: VOCAB/128 blocks of 256 threads (8 waves). Wave -> 16 vocab cols, 4 M-tiles.
__global__ __launch_bounds__(256)
void gemm_kernel(const float* __restrict__ x, const float* __restrict__ w,
                 float* __restrict__ logits) {
    __shared__ float xs[BATCH * LDS_STRIDE];

    const int tid  = threadIdx.x;
    const int wave = tid >> 5;
    const int lane = tid & 31;
    const int l16  = lane & 15;
    const int hi   = lane >> 4;            // 0 or 1 -> K offset 0 or 2
    const int vBase = blockIdx.x * 128 + wave * 16;
    const float* wrow = w + (size_t)(vBase + l16) * DIM + hi * 2;

    v8f acc[4];
#pragma unroll
    for (int mt = 0; mt < 4; ++mt) acc[mt] = (v8f)0.f;

    const int ldr = tid >> 2;              // 0..63  (batch row for coop load)
    const int ldc = (tid & 3) * 8;         // 0,8,16,24

    for (int k0 = 0; k0 < DIM; k0 += BK) {
        // cooperative stage x[0:64, k0:k0+BK] into LDS (8 floats/thread)
        {
            const v4f* src = (const v4f*)(x + ldr * DIM + k0 + ldc);
            v4f p0 = src[0], p1 = src[1];
            float* dst = xs + ldr * LDS_STRIDE + ldc;
            *(v4f*)(dst)     = p0;
            *(v4f*)(dst + 4) = p1;
        }
        __syncthreads();

        // prefetch next chunk of the weight stream (gfx1250 global_prefetch_b8)
        if (k0 + BK < DIM) __builtin_prefetch(wrow + k0 + BK, 0, 1);

        // preload all B fragments for this chunk: one 128B line per row pair
        v2f bf[BK / 4];
#pragma unroll
        for (int i = 0; i < BK / 4; ++i)
            bf[i] = *(const v2f*)(wrow + k0 + i * 4);

#pragma unroll
        for (int i = 0; i < BK / 4; ++i) {
            const int kk = i * 4;
#pragma unroll
            for (int mt = 0; mt < 4; ++mt) {
                v2f a = *(const v2f*)(xs + (mt * 16 + l16) * LDS_STRIDE + kk + hi * 2);
                acc[mt] = __builtin_amdgcn_wmma_f32_16x16x4_f32(
                    /*neg_a=*/false, a, /*neg_b=*/false, bf[i],
                    /*c_mod=*/(short)0, acc[mt],
                    /*reuse_a=*/false, /*reuse_b=*/false);
            }
        }
        __syncthreads();
    }

    // D layout: c[j] -> row M = j + hi*8 (+ mt*16), col N = vBase + l16
#pragma unroll
    for (int mt = 0; mt < 4; ++mt) {
#pragma unroll
        for (int j = 0; j < 8; ++j) {
            const int m = mt * 16 + j + hi * 8;
            logits[(size_t)m * VOCAB + vBase + l16] = acc[mt][j];
        }
    }
}

// ------------- Top-50 extraction + temperature/top-p sampling head -------------
__global__ __launch_bounds__(256)
void topk_kernel(float* __restrict__ logits, const float* __restrict__ top_p,
                 const float* __restrict__ temperature, float* __restrict__ out) {
    __shared__ float svals[256];
    __shared__ int   sidx[256];
    __shared__ float top_v[TOPK];
    __shared__ int   top_i[TOPK];

    const int row = blockIdx.x, tid = threadIdx.x;
    float* lrow = logits + (size_t)row * VOCAB;

    for (int it = 0; it < TOPK; ++it) {
        float best = -INFINITY; int bidx = 0;
        for (int v = tid; v < VOCAB; v += 256) {
            float f = lrow[v];
            if (f > best) { best = f; bidx = v; }
        }
        svals[tid] = best; sidx[tid] = bidx;
        __syncthreads();
        for (int st = 128; st > 0; st >>= 1) {
            if (tid < st && svals[tid + st] > svals[tid]) {
                svals[tid] = svals[tid + st]; sidx[tid] = sidx[tid + st];
            }
            __syncthreads();
        }
        if (tid == 0) {
            top_v[it] = svals[0]; top_i[it] = sidx[0];
            lrow[sidx[0]] = -INFINITY;      // remove winner for next pass
            __threadfence_block();
        }
        __syncthreads();
    }

    if (tid == 0) {
        const float invT = 1.0f / temperature[0];
        const float tp   = top_p[0];
        float l[TOPK], e[TOPK];
        float mx = -INFINITY;
        for (int i = 0; i < TOPK; ++i) { l[i] = top_v[i] * invT; mx = fmaxf(mx, l[i]); }
        float sum = 0.f;
        for (int i = 0; i < TOPK; ++i) { e[i] = __expf(l[i] - mx); sum += e[i]; }
        const float inv = 1.0f / sum;
        float cum = 0.f;
        for (int i = 0; i < TOPK; ++i) {
            cum += e[i] * inv;                       // cumulative prob incl. i
            const bool keep = (cum < tp) || (i < MINKEEP);
            l[i] = keep ? l[i] : -1000.0f;           // filtered logits
        }
        float mx2 = -INFINITY;
        for (int i = 0; i < TOPK; ++i) mx2 = fmaxf(mx2, l[i]);
        float sum2 = 0.f;
        for (int i = 0; i < TOPK; ++i) { e[i] = __expf(l[i] - mx2); sum2 += e[i]; }
        const float inv2 = 1.0f / sum2;
        float* probs = out + (size_t)row * TOPK;
        int*   toks  = (int*)out + (size_t)BATCH * TOPK + (size_t)row * TOPK;
        for (int i = 0; i < TOPK; ++i) {
            probs[i] = e[i] * inv2;
            toks[i]  = top_i[i];
        }
    }
}

extern "C" void kernel_launch(void* const* d_in, const int* in_sizes, int n_in,
                              void* d_out, int out_size, void* d_ws, size_t ws_size,
                              hipStream_t stream) {
    const float* hs   = (const float*)d_in[0];   // [64,4096]
    const float* lnw  = (const float*)d_in[1];   // [4096]
    const float* lnb  = (const float*)d_in[2];   // [4096]
    const float* lmw  = (const float*)d_in[3];   // [32000,4096]
    const float* tp   = (const float*)d_in[4];   // [1]
    const float* temp = (const float*)d_in[5];   // [1]

    float* xnorm  = (float*)d_ws;                       // 64*4096 f32 = 1 MB
    float* logits = xnorm + (size_t)BATCH * DIM;        // 64*32000 f32 = 8.2 MB

    ln_kernel  <<<BATCH,       256, 0, stream>>>(hs, lnw, lnb, xnorm);
    gemm_kernel<<<VOCAB / 128, 256, 0, stream>>>(xnorm, lmw, logits);
    topk_kernel<<<BATCH,       256, 0, stream>>>(logits, tp, temp, (float*)d_out);
}